// HighwayLSTMLayer_71640054497421
// MI455X (gfx1250) — compile-verified
//
#include <hip/hip_runtime.h>
#include <hip/hip_bf16.h>
#include <math.h>

// Problem constants
#define DD   512
#define HH   512
#define LL   3
#define BB   128
#define TT   256
#define G6   3072      // 6*H
#define G5   2560      // 5*H

typedef __attribute__((ext_vector_type(2))) float v2f;
typedef __attribute__((ext_vector_type(8))) float v8f;

// fp32 WMMA: D = A(16x4) * B(4x16) + C(16x16), wave32.
static __device__ __forceinline__ v8f wmma_f32(v2f a, v2f b, v8f c) {
  return __builtin_amdgcn_wmma_f32_16x16x4_f32(false, a, false, b, (short)0, c,
                                               false, false);
}

static __device__ __forceinline__ float sigmoidf_(float x) {
  return 1.0f / (1.0f + __expf(-x));
}

// -------------------------------------------------------------------------
// Kernel A: tmp_i[r, g] = sum_k X[r, k] * W_ih[k, g]
//   M = T*B = 32768 (row r = t*B + b), K = 512, N = 3072.
//   layer0_mode: X is x[B,T,D]; else X is hid[T,B,H] (row-major r*512+k).
// Block = 128 threads (4 waves), tile 64(M) x 64(N); wave w owns M-strip
// [w*16, w*16+16) x all 64 N cols => 4 accumulator tiles of 16x16.
// A staged as float4 (16B-aligned rows, pad 20 -> conflict-free fragment
// reads); B staged pair-interleaved so each wmma B fragment is one v2f read.
// -------------------------------------------------------------------------
__global__ __launch_bounds__(128) void gemm_ih_kernel(
    const float* __restrict__ X, const float* __restrict__ W,
    float* __restrict__ Cout, int layer0_mode) {
  __shared__ float As[64][20];    // 80B rows: 16B aligned; 20r mod 64 distinct
  __shared__ float Bs[8][128];    // pair-interleaved: [k/2][col*2 + (k&1)]

  const int bm    = blockIdx.x;   // M/64 = 512
  const int bn    = blockIdx.y;   // N/64 = 48
  const int tid   = threadIdx.x;
  const int wave  = tid >> 5;
  const int lane  = tid & 31;
  const int lrow  = lane & 15;
  const int lhalf = lane >> 4;    // 0: K pair (k0,k0+1), 1: (k0+2,k0+3)

  v8f acc[4] = {};

  for (int kc = 0; kc < 512; kc += 16) {
    // Stage A chunk: 64 rows x 16 k, float4 per thread x2
    for (int i = tid; i < 256; i += 128) {    // 256 float4 slots
      const int r = i >> 2, c4 = (i & 3) * 4;
      const int row = bm * 64 + r;
      size_t addr;
      if (layer0_mode) {
        const int t = row >> 7;               // row / B
        const int b = row & 127;              // row % B
        addr = (size_t)b * (TT * DD) + (size_t)t * DD + (kc + c4);
      } else {
        addr = (size_t)row * 512 + (kc + c4);
      }
      *(float4*)&As[r][c4] = *(const float4*)&X[addr];
    }
    // Stage B chunk pair-interleaved: 8 pair-rows x 64 cols, v2f per slot
    for (int i = tid; i < 512; i += 128) {    // 512 v2f slots
      const int pr = i >> 6, col = i & 63;
      const size_t g0 = (size_t)(kc + 2 * pr) * G6 + bn * 64 + col;
      v2f b2;
      b2.x = W[g0];
      b2.y = W[g0 + G6];
      *(v2f*)&Bs[pr][col * 2] = b2;
    }
    // Prefetch next chunk while this one is consumed
    if (kc + 16 < 512) {
      {  // B: 16 rows x 64 cols, 32B granularity coverage
        const int r = tid >> 3, c = (tid & 7) * 8;
        __builtin_prefetch(&W[(size_t)(kc + 16 + r) * G6 + bn * 64 + c], 0, 0);
      }
      {  // A: 64 rows x 16 cols, 32B granularity coverage
        const int r = tid >> 1, c = (tid & 1) * 8;
        const int row = bm * 64 + r;
        size_t addr;
        if (layer0_mode) {
          const int t = row >> 7, b = row & 127;
          addr = (size_t)b * (TT * DD) + (size_t)t * DD + (kc + 16 + c);
        } else {
          addr = (size_t)row * 512 + (kc + 16 + c);
        }
        __builtin_prefetch(&X[addr], 0, 0);
      }
    }
    __syncthreads();

    #pragma unroll
    for (int k0 = 0; k0 < 16; k0 += 4) {
      const v2f af = *(const v2f*)&As[wave * 16 + lrow][k0 + lhalf * 2];
      const int pr = (k0 >> 1) + lhalf;       // (k0 + lhalf*2) / 2
      #pragma unroll
      for (int n = 0; n < 4; ++n) {
        const v2f bf = *(const v2f*)&Bs[pr][(n * 16 + lrow) * 2];
        acc[n] = wmma_f32(af, bf, acc[n]);
      }
    }
    __syncthreads();
  }

  // Store: C/D layout -> M = v + lhalf*8, N = lrow
  #pragma unroll
  for (int n = 0; n < 4; ++n) {
    #pragma unroll
    for (int v = 0; v < 8; ++v) {
      const int row = bm * 64 + wave * 16 + v + lhalf * 8;
      const int col = bn * 64 + n * 16 + lrow;
      Cout[(size_t)row * G6 + col] = acc[n][v];
    }
  }
}

// -------------------------------------------------------------------------
// Repack W_hh [512, 2560] -> Wp pair-interleaved: Wp[k/2][col][2].
// Lets each step-kernel B fragment be a single coalesced global b64 load.
// -------------------------------------------------------------------------
__global__ __launch_bounds__(256) void repack_whh_kernel(
    const float* __restrict__ Whh, float* __restrict__ Wp) {
  const int i = blockIdx.x * blockDim.x + threadIdx.x;   // 512*2560 elems
  if (i < 512 * G5) {
    const int k = i / G5;
    const int c = i - k * G5;
    Wp[(size_t)(k >> 1) * (2 * G5) + c * 2 + (k & 1)] = Whh[i];
  }
}

// -------------------------------------------------------------------------
// Kernel B: one scan step of the highway LSTM for one layer.
// Grid (8 btiles, 8 jtiles), block = 128 threads (4 waves).
// Block stages h_drop_in[btile*16 .. +16][0..512) in LDS; each wave computes
// one 16x16 j-tile for all 5 gates (th) from the pair-interleaved W_hh, then
// fuses gates + highway + per-batch length masking (reversed-time gather of
// tmp_i for odd layers). h double-buffered across steps; c elementwise
// in-place (single owner per (b,j)).
// -------------------------------------------------------------------------
__global__ __launch_bounds__(128) void lstm_step_kernel(
    const float* __restrict__ ti,      // tmp_i [T,B,3072] (unflipped time)
    const float* __restrict__ Wp,      // [256][2560][2] pair-interleaved W_hh
    const float* __restrict__ bias,    // [2560]
    const int*   __restrict__ lengths, // [B]
    const float* __restrict__ drop,    // [B,512] this layer
    const float* __restrict__ hin,     // [B,512] h*drop, previous step
    float*       __restrict__ hout,    // [B,512] h*drop, this step
    float*       __restrict__ cstate,  // [B,512]
    float*       __restrict__ hid_out, // [T,B,512] this layer
    int s, int reverse) {
  __shared__ float As[16][516];        // 2064B rows: 16B aligned; 4r mod 64 ok

  const int btile = blockIdx.x;        // 8
  const int jtile = blockIdx.y;        // 8
  const int tid   = threadIdx.x;
  const int wave  = tid >> 5;
  const int lane  = tid & 31;
  const int lrow  = lane & 15;
  const int lhalf = lane >> 4;

  // Stage A: 16 batch rows x 512 K, float4 loads/stores
  for (int i = tid; i < 2048; i += 128) {        // 2048 float4 slots
    const int r = i >> 7, c4 = (i & 127) * 4;
    *(float4*)&As[r][c4] =
        *(const float4*)&hin[(size_t)(btile * 16 + r) * 512 + c4];
  }
  __syncthreads();

  v8f acc[5] = {};
  const int colbase = jtile * 64 + wave * 16 + lrow;   // j within H
  const float* __restrict__ Wb = Wp + (size_t)colbase * 2;

  #pragma unroll 2
  for (int k0 = 0; k0 < 512; k0 += 4) {
    const v2f af = *(const v2f*)&As[lrow][k0 + lhalf * 2];
    const size_t prow = (size_t)((k0 >> 1) + lhalf) * (2 * G5);
    #pragma unroll
    for (int g = 0; g < 5; ++g) {
      const v2f bf = *(const v2f*)&Wb[prow + g * 1024];
      acc[g] = wmma_f32(af, bf, acc[g]);
    }
  }

  // Fused gate epilogue
  const int j = colbase;
  #pragma unroll
  for (int v = 0; v < 8; ++v) {
    const int b = btile * 16 + v + lhalf * 8;
    const int len = lengths[b];
    const bool valid = (s < len);
    if (valid) {
      const int t = reverse ? (len - 1 - s) : s;
      const size_t tirow = ((size_t)t * BB + b) * (size_t)G6;
      const float g0 = acc[0][v] + ti[tirow + 0 * 512 + j] + bias[0 * 512 + j];
      const float g1 = acc[1][v] + ti[tirow + 1 * 512 + j] + bias[1 * 512 + j];
      const float g2 = acc[2][v] + ti[tirow + 2 * 512 + j] + bias[2 * 512 + j];
      const float g3 = acc[3][v] + ti[tirow + 3 * 512 + j] + bias[3 * 512 + j];
      const float g4 = acc[4][v] + ti[tirow + 4 * 512 + j] + bias[4 * 512 + j];
      const float lin = ti[tirow + 5 * 512 + j];
      const float ig = sigmoidf_(g0);
      const float fg = sigmoidf_(g1);
      const float gc = tanhf(g2);
      const float og = sigmoidf_(g3);
      const float rg = sigmoidf_(g4);
      const size_t bj = (size_t)b * 512 + j;
      const float cnew = fg * cstate[bj] + ig * gc;
      const float hnew = rg * (og * tanhf(cnew)) + (1.0f - rg) * lin;
      cstate[bj] = cnew;
      hout[bj] = hnew * drop[bj];
      hid_out[((size_t)t * BB + b) * 512 + j] = hnew;
    } else {
      // h carry becomes 0 and stays; c frozen; output position s is zero
      // (for reversed layers, invalid flipped index s maps back to s).
      hout[(size_t)b * 512 + j] = 0.0f;
      hid_out[((size_t)s * BB + b) * 512 + j] = 0.0f;
    }
  }
}

__global__ void zero_kernel(float* __restrict__ p, int n) {
  const int i = blockIdx.x * blockDim.x + threadIdx.x;
  if (i < n) p[i] = 0.0f;
}

// output[b,t,j] = hid2[t,b,j]
__global__ __launch_bounds__(128) void transpose_out_kernel(
    const float* __restrict__ hid2, float* __restrict__ out) {
  const int t = blockIdx.x;
  const int b = blockIdx.y;
  const float4* src = (const float4*)(hid2 + ((size_t)t * BB + b) * 512);
  float4* dst = (float4*)(out + ((size_t)b * TT + t) * 512);
  dst[threadIdx.x] = src[threadIdx.x];
}

extern "C" void kernel_launch(void* const* d_in, const int* in_sizes, int n_in,
                              void* d_out, int out_size, void* d_ws,
                              size_t ws_size, hipStream_t stream) {
  (void)in_sizes; (void)n_in; (void)out_size; (void)ws_size;
  const float* x       = (const float*)d_in[0];  // [B,T,D]
  const int*   lengths = (const int*)d_in[1];    // [B]
  const float* weight  = (const float*)d_in[2];  // flat
  const float* bias    = (const float*)d_in[3];  // [3*5*H]
  const float* dropout = (const float*)d_in[4];  // [L,B,H]

  float* out = (float*)d_out;                         // [B,T,H]
  float* hid = out + (size_t)BB * TT * HH;            // [L,T,B,H]

  float* ws     = (float*)d_ws;
  float* tmp_i  = ws;                                 // [T,B,6H] = 100663296
  float* hbuf0  = ws + (size_t)100663296;             // [B,H]
  float* hbuf1  = hbuf0 + (size_t)BB * HH;            // [B,H]
  float* cbuf   = hbuf1 + (size_t)BB * HH;            // [B,H]
  float* wpbuf  = cbuf + (size_t)BB * HH;             // [256][2560][2]

  // float offsets into `weight` (in_l = 512 for all layers)
  const size_t wih_off[3] = {0, 2883584, 5767168};
  const size_t whh_off[3] = {1572864, 4456448, 7340032};
  const size_t layer_hid = (size_t)TT * BB * HH;      // 16777216

  for (int l = 0; l < LL; ++l) {
    const float* X = (l == 0) ? x : (hid + (size_t)(l - 1) * layer_hid);

    dim3 gA(512, 48);   // (M/64, N/64)
    gemm_ih_kernel<<<gA, 128, 0, stream>>>(X, weight + wih_off[l], tmp_i,
                                           l == 0 ? 1 : 0);

    // Pair-interleave this layer's W_hh for b64 fragment loads
    const int nrp = 512 * G5;
    repack_whh_kernel<<<(nrp + 255) / 256, 256, 0, stream>>>(
        weight + whh_off[l], wpbuf);

    // zero hbuf0, hbuf1, cbuf (contiguous 3*B*H floats)
    const int nz = 3 * BB * HH;
    zero_kernel<<<(nz + 255) / 256, 256, 0, stream>>>(hbuf0, nz);

    const int reverse = (l & 1);
    float* hl = hid + (size_t)l * layer_hid;
    for (int s = 0; s < TT; ++s) {
      float* hin  = (s & 1) ? hbuf1 : hbuf0;
      float* hout = (s & 1) ? hbuf0 : hbuf1;
      lstm_step_kernel<<<dim3(8, 8), 128, 0, stream>>>(
          tmp_i, wpbuf, bias + (size_t)l * G5, lengths,
          dropout + (size_t)l * BB * HH, hin, hout, cbuf, hl, s, reverse);
    }
  }

  transpose_out_kernel<<<dim3(TT, BB), 128, 0, stream>>>(
      hid + (size_t)2 * layer_hid, out);
}